// CPN_15693810499943
// MI455X (gfx1250) — compile-verified
//
#include <hip/hip_runtime.h>
#include <hip/hip_bf16.h>
#include <stdint.h>

typedef __attribute__((ext_vector_type(2))) float v2f;
typedef __attribute__((ext_vector_type(8))) float v8f;

#define K_UNITS 4096
#define TILE_N  16
#define BLOCK_THREADS 256
#define WAVES_PER_BLOCK 8   // 256 / wave32
#define ROWS_PER_WAVE 16

__global__ __launch_bounds__(BLOCK_THREADS)
void cpn_wmma_kernel(const float* __restrict__ x,      // [B,2]
                     const float* __restrict__ koh,    // [K,3] (row-normalized)
                     const float* __restrict__ gross,  // [K,1]
                     float* __restrict__ out,          // [B,1]
                     int B)
{
    // Stage kohonen weights + ||w||^2 into LDS: 4096 * float4 = 64 KB (<= 320 KB/WGP)
    __shared__ float4 w_lds[K_UNITS];

    const int tid = threadIdx.x;
    #pragma unroll
    for (int it = 0; it < K_UNITS / BLOCK_THREADS; ++it) {
        int k = tid + it * BLOCK_THREADS;
        float w0 = koh[3 * k + 0];
        float w1 = koh[3 * k + 1];
        float w2 = koh[3 * k + 2];
        float wb = w0 * w0 + w1 * w1 + w2 * w2;
        w_lds[k] = make_float4(w0, w1, w2, wb);
    }
    __syncthreads();

    const int lane    = tid & 31;
    const int wave    = tid >> 5;
    const int rowBase = (blockIdx.x * WAVES_PER_BLOCK + wave) * ROWS_PER_WAVE;
    const bool hi     = (lane >= 16);        // lanes 16..31 carry K=2,3 slices
    const int  sub    = lane & 15;

    // ---- preprocess: project x onto upper unit hemisphere (all 32 lanes compute
    // the point for row (rowBase + sub); hi half duplicates lo half's work) ----
    int p = rowBase + sub;
    float x0 = x[2 * p + 0];
    float x1 = x[2 * p + 1];
    float radial = sqrtf(x0 * x0 + x1 * x1);
    float inv    = 1.0f / fmaxf(radial, 1.0f);
    float xs0 = x0 * inv;
    float xs1 = x1 * inv;
    float r2  = xs0 * xs0 + xs1 * xs1;
    float x3  = sqrtf(fmaxf(fminf(1.0f - r2, 1.0f), 0.0f));

    // ---- A matrix (16x4 f32): K-vector per row = (-2*xs0, -2*xs1, -2*x3, 1.0)
    // so D = A x B + 0 = wb - 2*(xp . w) = d2   (wb rides in B's K=3 row).
    // Layout: VGPR0: lanes0-15 K=0, lanes16-31 K=2; VGPR1: K=1 / K=3.
    v2f A;
    A.x = hi ? (-2.0f * x3) : (-2.0f * xs0);
    A.y = hi ? 1.0f         : (-2.0f * xs1);

    float bestMin[8];
    int   bestIdx[8];
    #pragma unroll
    for (int i = 0; i < 8; ++i) { bestMin[i] = __int_as_float(0x7f800000); bestIdx[i] = 0x7fffffff; }

    // ---- sweep K in 16-wide tiles, software-pipelined LDS prefetch ----
    int    col = sub;
    float4 w   = w_lds[col];

    #pragma unroll 4
    for (int n0 = 0; n0 < K_UNITS; n0 += TILE_N) {
        float4 wcur = w;
        int    ccur = col;

        // branchless prefetch of next tile (wraps to 0 on last iteration)
        int nnext = (n0 + TILE_N) & (K_UNITS - 1);
        col = nnext + sub;
        w   = w_lds[col];

        // B matrix (4x16 f32): VGPR0: K=0 row (w0) / K=2 row (w2);
        //                      VGPR1: K=1 row (w1) / K=3 row (wb)
        v2f Bm;
        Bm.x = hi ? wcur.z : wcur.x;
        Bm.y = hi ? wcur.w : wcur.y;

        v8f C = {};  // inline 0 accumulator
        v8f D = __builtin_amdgcn_wmma_f32_16x16x4_f32(
            /*neg_a=*/false, A, /*neg_b=*/false, Bm,
            /*c_mod=*/(short)0, C, /*reuse_a=*/false, /*reuse_b=*/false);

        #pragma unroll
        for (int i = 0; i < 8; ++i) {
            float d = D[i];
            if (d < bestMin[i]) { bestMin[i] = d; bestIdx[i] = ccur; }  // strict < keeps lowest idx per lane
        }
    }

    // ---- branchless cross-lane argmin: pack (sortable(min) << 32 | idx), u64-min
    // over XOR shuffles 1,2,4,8 (stays inside each 16-lane half; halves own
    // disjoint row sets). Low-bits index => ties resolve to the smaller index.
    #pragma unroll
    for (int i = 0; i < 8; ++i) {
        uint32_t fb  = __float_as_uint(bestMin[i]);
        uint32_t khi = (fb & 0x80000000u) ? ~fb : (fb | 0x80000000u);  // order-preserving map
        uint32_t klo = (uint32_t)bestIdx[i];

        #pragma unroll
        for (int off = 1; off < 16; off <<= 1) {
            uint32_t ohi = (uint32_t)__shfl_xor((int)khi, off, 32);
            uint32_t olo = (uint32_t)__shfl_xor((int)klo, off, 32);
            uint64_t mine  = ((uint64_t)khi << 32) | klo;
            uint64_t other = ((uint64_t)ohi << 32) | olo;
            uint64_t m = (other < mine) ? other : mine;
            khi = (uint32_t)(m >> 32);
            klo = (uint32_t)m;
        }
        bestIdx[i] = (int)klo;
    }

    // ---- gather grossberg weight of winner; lane0 writes rows 0..7, lane16 rows 8..15 ----
    if (sub == 0) {
        int mbase = rowBase + (hi ? 8 : 0);
        #pragma unroll
        for (int i = 0; i < 8; ++i) {
            out[mbase + i] = gross[bestIdx[i]];
        }
    }
}

extern "C" void kernel_launch(void* const* d_in, const int* in_sizes, int n_in,
                              void* d_out, int out_size, void* d_ws, size_t ws_size,
                              hipStream_t stream) {
    (void)n_in; (void)d_ws; (void)ws_size; (void)out_size;
    const float* x     = (const float*)d_in[0];   // [B,2]
    const float* koh   = (const float*)d_in[1];   // [4096,3]
    const float* gross = (const float*)d_in[2];   // [4096,1]
    float*       out   = (float*)d_out;           // [B,1]

    int B = in_sizes[0] / 2;                      // 65536
    int rowsPerBlock = WAVES_PER_BLOCK * ROWS_PER_WAVE;  // 128
    int blocks = (B + rowsPerBlock - 1) / rowsPerBlock;  // 512

    cpn_wmma_kernel<<<blocks, BLOCK_THREADS, 0, stream>>>(x, koh, gross, out, B);
}